// ModelNew_63806034150019
// MI455X (gfx1250) — compile-verified
//
#include <hip/hip_runtime.h>
#include <hip/hip_bf16.h>

// Problem constants: x (16,32,32,32,32) f32, folded g (8 taps,32 ci,64 co),
// out (16,64,31,31,31) f32.
#define NB    16
#define CI    32
#define CO    64
#define DIN   32
#define DOUTD 31

typedef __attribute__((ext_vector_type(2))) float v2f;
typedef __attribute__((ext_vector_type(8))) float v8f;
typedef unsigned int uint32x4 __attribute__((ext_vector_type(4)));
typedef int          int32x8  __attribute__((ext_vector_type(8)));
typedef int          int32x4  __attribute__((ext_vector_type(4)));

// LDS layout (floats), dynamic shared:
//   gs2: 8*8*2 rows x 80 float2 (pair-interleaved, bank-padded) = 20480 floats
//   xs : 128 rows x 33 (TDM pad col)                            =  4224 floats
//   ob : 64 x 31 output tile (contiguous for TDM store)         =  1984 floats
#define GS_F (8 * 8 * 2 * 80 * 2)
#define XS_F (128 * 33)
#define OB_F (CO * DOUTD)

// ---------------------------------------------------------------------------
// Tensor Data Mover descriptor helpers (CDNA5 D#, CDNA5 ISA).
// This toolchain's builtin arity is 6:
//   (uint32x4 g0, int32x8 g1, int32x4 g2, int32x4 g3, int32x8 extra, i32 cpol)
// The ISA D# is exactly groups 0-3; the trailing int32x8 is passed zeroed.
// ---------------------------------------------------------------------------
__device__ __forceinline__ uint32x4 tdm_group0(uint32_t lds_addr, const void* gptr) {
    uint64_t ga = (uint64_t)(uintptr_t)gptr;
    uint32x4 g0;
    g0.x = 1u;                                              // count=1, user desc
    g0.y = lds_addr;                                        // LDS byte address
    g0.z = (uint32_t)ga;                                    // global_addr[31:0]
    g0.w = (uint32_t)((ga >> 32) & 0x01FFFFFFu) | (2u << 30); // addr[56:32], type=2
    return g0;
}

__device__ __forceinline__ int32x8 tdm_group1(uint32_t tensor_d0, uint32_t tensor_d1,
                                              uint32_t tile_d0, uint32_t tile_d1,
                                              uint32_t tile_d2,
                                              uint64_t stride0, uint64_t stride1,
                                              uint32_t pad_en, uint32_t pad_interval,
                                              uint32_t pad_amount) {
    int32x8 g1;
    g1[0] = (int)((2u << 16) |                 // data_size = 4 bytes
                  (pad_en << 20) | (pad_interval << 22) | (pad_amount << 25));
    g1[1] = (int)((tensor_d0 & 0xFFFFu) << 16);
    g1[2] = (int)(((tensor_d0 >> 16) & 0xFFFFu) | ((tensor_d1 & 0xFFFFu) << 16));
    g1[3] = (int)(((tensor_d1 >> 16) & 0xFFFFu) | (tile_d0 << 16));
    g1[4] = (int)(tile_d1 | (tile_d2 << 16));
    g1[5] = (int)(stride0 & 0xFFFFFFFFu);
    g1[6] = (int)(((stride0 >> 32) & 0xFFFFu) | ((stride1 & 0xFFFFu) << 16));
    g1[7] = (int)((stride1 >> 16) & 0xFFFFFFFFu);
    return g1;
}

__device__ __forceinline__ int32x4 tdm_group2(uint32_t tensor_d2, uint32_t tensor_d3,
                                              uint64_t stride2, uint32_t tile_d3) {
    int32x4 g2;
    g2[0] = (int)tensor_d2;
    g2[1] = (int)tensor_d3;
    g2[2] = (int)(stride2 & 0xFFFFFFFFu);
    g2[3] = (int)(((stride2 >> 32) & 0xFFFFu) | (tile_d3 << 16));
    return g2;
}

// ---------------------------------------------------------------------------
// Prep: fold ConvTranspose3d(k=3,s=2,p=1)+AvgPool(2) weight into 2x2x2 taps,
// written directly in the WMMA-B pair-interleaved order consumed by the main
// kernel: flat index = (((tap*8 + kk)*2 + hi)*64 + co)*2 + parity,
// with ci = kk*4 + hi*2 + parity.  S(0)={1,2}, S(1)={0} per spatial dim.
// ---------------------------------------------------------------------------
__global__ __launch_bounds__(256)
void fold_weights_kernel(const float* __restrict__ w,
                         const float* __restrict__ bias_conv,
                         const float* __restrict__ s1p,
                         const float* __restrict__ biasp,
                         const float* __restrict__ s2p,
                         float* __restrict__ g,
                         float* __restrict__ bias_eff,
                         float* __restrict__ factorp) {
    int tid = blockIdx.x * blockDim.x + threadIdx.x;
    if (tid < 8 * CI * CO) {
        int parity = tid & 1;
        int co     = (tid >> 1) & 63;
        int hi     = (tid >> 7) & 1;
        int kk     = (tid >> 8) & 7;
        int tap    = tid >> 11;
        int ci     = kk * 4 + hi * 2 + parity;
        int td = (tap >> 2) & 1, th = (tap >> 1) & 1, tw = tap & 1;
        const float* wp = w + (ci * CO + co) * 27;
        float s = 0.f;
        #pragma unroll
        for (int kd = 0; kd < 3; ++kd) {
            if (td ? (kd != 0) : (kd == 0)) continue;
            #pragma unroll
            for (int kh = 0; kh < 3; ++kh) {
                if (th ? (kh != 0) : (kh == 0)) continue;
                #pragma unroll
                for (int kw = 0; kw < 3; ++kw) {
                    if (tw ? (kw != 0) : (kw == 0)) continue;
                    s += wp[kd * 9 + kh * 3 + kw];
                }
            }
        }
        g[tid] = s;
    }
    if (tid < CO) {
        float s1 = s1p[0], s2 = s2p[0];
        bias_eff[tid] = (bias_conv[tid] * s1 + biasp[tid]) * s2;
        if (tid == 0) factorp[0] = s1 * s2 * 0.125f;
    }
}

// ---------------------------------------------------------------------------
// Main: implicit GEMM, V_WMMA_F32_16X16X4_F32 at full f32 precision.
// One block per (n,d,h): out[n, 0..63, d, h, 0..30].
// M = w (31 -> 32), N = co (64), K = ci*8taps (256); 8 waves = 2 mtile x 4 cotile.
// All tile movement via Tensor Data Mover; epilogue store via TDM 2D store.
// ---------------------------------------------------------------------------
__global__ __launch_bounds__(256)
void conv_fold_wmma_kernel(const float* __restrict__ x,
                           const float* __restrict__ g,
                           const float* __restrict__ bias_eff,
                           const float* __restrict__ factorp,
                           float* __restrict__ out) {
    extern __shared__ float smem[];
    float* gs = smem;               // pair-interleaved B, rows padded to 80 f2
    float* xs = smem + GS_F;        // [ci][dd][hh][w] rows of 32 + 1 pad
    float* ob = smem + GS_F + XS_F; // [co][31] contiguous

    const int tid  = threadIdx.x;
    const int h    = blockIdx.x;    // 0..30
    const int d    = blockIdx.y;    // 0..30
    const int n    = blockIdx.z;    // 0..15
    const int lane = tid & 31;
    const int wave = tid >> 5;
    const int mtile = wave >> 2;    // 0..1  -> w base 0/16
    const int ctile = wave & 3;     // 0..3  -> co base 0/16/32/48
    const int l16  = lane & 15;
    const int hi   = lane >> 4;
    const int cb   = ctile * 16 + l16;   // this lane's co column

    const float factor = factorp[0];
    const float be     = bias_eff[cb];

    const uint32_t lds0 = (uint32_t)(uintptr_t)smem;
    const int32x4 gz4 = {0, 0, 0, 0};
    const int32x8 gz8 = {0, 0, 0, 0, 0, 0, 0, 0};

    if (wave == 0) {
        // --- TDM #1: folded weights, 16384 dwords 1D, pad 32 DW per 128 DW
        //     -> LDS rows of 80 float2 (bank-pair disjoint for hi=0/1). ---
        __builtin_amdgcn_tensor_load_to_lds(
            tdm_group0(lds0, g),
            tdm_group1(/*td0*/16384, /*td1*/0, /*tile0*/16384, /*tile1*/0, /*tile2*/0,
                       /*s0*/0, /*s1*/0, /*pad*/1, /*intv 128DW*/6, /*amt 32DW*/31),
            gz4, gz4, gz8, 0);
        // --- TDM #2: x slab x[n, 0..31, d..d+1, h..h+1, 0..31] as 4D tile
        //     (w,hh,dd,ci), pad 1 DW per 32 DW row -> stride-33 rows. ---
        const float* xbase =
            x + ((((size_t)n * CI) * DIN + d) * DIN + h) * DIN;
        __builtin_amdgcn_tensor_load_to_lds(
            tdm_group0(lds0 + GS_F * 4, xbase),
            tdm_group1(/*td0*/DIN, /*td1*/DIN, /*tile0*/32, /*tile1*/2, /*tile2*/2,
                       /*s0 hh*/DIN, /*s1 dd*/DIN * DIN,
                       /*pad*/1, /*intv 32DW*/4, /*amt 1DW*/0),
            tdm_group2(/*td2*/DIN, /*td3*/CI, /*s2 ci*/DIN * DIN * DIN, /*tile3*/CI),
            gz4, gz8, 0);
        __builtin_amdgcn_s_wait_tensorcnt(0);
    }
    __syncthreads();

    // ---- 64 x V_WMMA_F32_16X16X4_F32 ----
    v8f acc = {0.f, 0.f, 0.f, 0.f, 0.f, 0.f, 0.f, 0.f};
    #pragma unroll
    for (int tap = 0; tap < 8; ++tap) {
        const int td = tap >> 2, th = (tap >> 1) & 1, tw = tap & 1;
        const int wA = mtile * 16 + l16 + tw;   // <= 32 (pad col, masked rows only)
        const int rowbase = td * 2 + th;
        #pragma unroll
        for (int kk = 0; kk < 8; ++kk) {
            const int ciA = kk * 4 + hi * 2;    // this lane's K pair (ciA, ciA+1)
            v2f a, b;
            a.x = xs[(ciA * 4 + rowbase) * 33 + wA];
            a.y = xs[((ciA + 1) * 4 + rowbase) * 33 + wA];
            b = *(const v2f*)(gs + (((tap * 8 + kk) * 2 + hi) * 80 + cb) * 2);
            acc = __builtin_amdgcn_wmma_f32_16x16x4_f32(
                false, a, false, b, (short)0, acc, false, false);
        }
    }

    // ---- Fused epilogue into contiguous [co][31] LDS tile ----
    // C/D layout: VGPR r -> M = r (lanes 0-15) / r+8 (lanes 16-31); N = lane&15.
    #pragma unroll
    for (int r = 0; r < 8; ++r) {
        int m = mtile * 16 + r + hi * 8;
        if (m < DOUTD)
            ob[cb * DOUTD + m] = acc[r] * factor + be;
    }
    __syncthreads();

    if (wave == 0) {
        // --- TDM #3: 2D store, 64 rows of 31 dwords, row stride 31^3. ---
        float* obase = out + (size_t)n * CO * (DOUTD * DOUTD * DOUTD)
                           + (size_t)d * (DOUTD * DOUTD) + (size_t)h * DOUTD;
        __builtin_amdgcn_tensor_store_from_lds(
            tdm_group0(lds0 + (GS_F + XS_F) * 4, obase),
            tdm_group1(/*td0*/DOUTD, /*td1*/CO, /*tile0*/DOUTD, /*tile1*/CO, /*tile2*/0,
                       /*s0 co*/(uint64_t)DOUTD * DOUTD * DOUTD, /*s1*/0,
                       /*pad*/0, 0, 0),
            gz4, gz4, gz8, 0);
        // S_ENDPGM performs an implicit wait-idle (covers TENSORcnt).
    }
}

// ---------------------------------------------------------------------------
// Launch
// ---------------------------------------------------------------------------
extern "C" void kernel_launch(void* const* d_in, const int* in_sizes, int n_in,
                              void* d_out, int out_size, void* d_ws, size_t ws_size,
                              hipStream_t stream) {
    const float* x         = (const float*)d_in[0];
    const float* weight    = (const float*)d_in[1];
    const float* bias_conv = (const float*)d_in[2];
    const float* scale1    = (const float*)d_in[3];
    const float* bias      = (const float*)d_in[4];
    const float* scale2    = (const float*)d_in[5];
    float* out = (float*)d_out;

    float* g        = (float*)d_ws;          // 16384 floats, permuted B layout
    float* bias_eff = g + 8 * CI * CO;       // 64
    float* factorp  = bias_eff + CO;         // 1

    fold_weights_kernel<<<64, 256, 0, stream>>>(weight, bias_conv, scale1,
                                                bias, scale2, g, bias_eff, factorp);

    dim3 grid(DOUTD, DOUTD, NB);             // (h, d, n)
    size_t lds_bytes = (size_t)(GS_F + XS_F + OB_F) * sizeof(float); // ~104 KB
    conv_fold_wmma_kernel<<<grid, 256, lds_bytes, stream>>>(x, g, bias_eff,
                                                            factorp, out);
}